// Quantizer_80599356277525
// MI455X (gfx1250) — compile-verified
//
#include <hip/hip_runtime.h>

// Quantizer: q = nearest(levels, tanh(x)), levels = linspace(-1,1,64).
// Memory-bound streaming kernel: 48 MB total traffic -> ~2.1 us at 23.3 TB/s.
// Uniform levels => argmin over 64 candidates collapses to a closed-form
// round+clamp; WMMA would only inflate work, so the optimal kernel is a
// B128-vectorized wave32 stream with v_tanh_f32 + global_prefetch_b8.

__device__ __forceinline__ float fast_tanh(float x) {
#if __has_builtin(__builtin_amdgcn_tanhf)
  return __builtin_amdgcn_tanhf(x);   // gfx1250 v_tanh_f32 (single trans op)
#elif __has_builtin(__builtin_amdgcn_tanh_f32)
  return __builtin_amdgcn_tanh_f32(x);
#else
  return tanhf(x);
#endif
}

__device__ __forceinline__ void quant1(float x, const float* __restrict__ levels,
                                       float& q, int& idx) {
  const float kHalfScale = 31.5f;  // (L-1)/2
  float t = fast_tanh(x);
  // nearest uniform level: round((t - (-1)) / step), step = 2/63
  float r = rintf((t + 1.0f) * kHalfScale);
  r = fminf(fmaxf(r, 0.0f), 63.0f);
  idx = (int)r;
  q = __ldg(levels + idx);  // bit-exact match with jnp.linspace values; 256B table lives in WGP$
}

__global__ __launch_bounds__(256) void quantizer_tanh64(
    const float* __restrict__ x, const float* __restrict__ levels,
    float* __restrict__ q_out, int* __restrict__ idx_out, int n4) {
  int i = blockIdx.x * blockDim.x + threadIdx.x;
  const int stride = gridDim.x * blockDim.x;
  for (; i < n4; i += stride) {
    // Prefetch the next grid-stride chunk (gfx1250 global_prefetch_b8;
    // speculative — silently dropped past the end of the buffer).
    __builtin_prefetch(x + (size_t)(i + stride) * 4, 0, 1);

    float4 v = reinterpret_cast<const float4*>(x)[i];  // global_load_b128

    float4 q;
    int4 id;
    quant1(v.x, levels, q.x, id.x);
    quant1(v.y, levels, q.y, id.y);
    quant1(v.z, levels, q.z, id.z);
    quant1(v.w, levels, q.w, id.w);

    reinterpret_cast<float4*>(q_out)[i] = q;    // global_store_b128
    reinterpret_cast<int4*>(idx_out)[i] = id;   // global_store_b128
  }
}

extern "C" void kernel_launch(void* const* d_in, const int* in_sizes, int n_in,
                              void* d_out, int out_size, void* d_ws, size_t ws_size,
                              hipStream_t stream) {
  (void)n_in; (void)d_ws; (void)ws_size; (void)out_size;

  const float* x      = (const float*)d_in[0];  // [N] float32
  const float* levels = (const float*)d_in[1];  // [64] float32
  const int n = in_sizes[0];                    // 4194304
  const int n4 = n >> 2;                        // float4 chunks

  // Output layout: [0, n)      -> quantized values (float32)
  //                [n, 2n)     -> argmin indices (int32)
  float* q_out  = (float*)d_out;
  int*   id_out = (int*)d_out + n;

  // ~0.5M threads in flight (2 float4 chunks each): enough waves to saturate
  // 23.3 TB/s while giving the prefetch a useful next chunk.
  const int block = 256;                 // 8 wave32 waves per block
  int grid = (n4 + block * 2 - 1) / (block * 2);
  if (grid < 1) grid = 1;

  quantizer_tanh64<<<grid, block, 0, stream>>>(x, levels, q_out, id_out, n4);
}